// baseGNN_30064771072231
// MI455X (gfx1250) — compile-verified
//
#include <hip/hip_runtime.h>
#include <hip/hip_bf16.h>

typedef float v2f __attribute__((ext_vector_type(2)));
typedef float v8f __attribute__((ext_vector_type(8)));

#define N_NODES 100000
#define N_EDGES 1600000
#define D 64

// ---------------------------------------------------------------------------
// zero fill
// ---------------------------------------------------------------------------
__global__ void zero_f32(float* __restrict__ p, int n) {
    int t = blockIdx.x * blockDim.x + threadIdx.x;
    if (t < n) p[t] = 0.0f;
}

// ---------------------------------------------------------------------------
// degree: deg[dst[e]] += 1  (float atomics, exact up to 2^24)
// ---------------------------------------------------------------------------
__global__ void degree_kernel(const int* __restrict__ dst, float* __restrict__ deg, int n_edges) {
    int e = blockIdx.x * blockDim.x + threadIdx.x;
    if (e < n_edges) atomicAdd(&deg[dst[e]], 1.0f);
}

// dinv[i] = rsqrt(deg[i] + 1)   (self-loop included)
__global__ void dinv_kernel(const float* __restrict__ deg, float* __restrict__ dinv, int n) {
    int t = blockIdx.x * blockDim.x + threadIdx.x;
    if (t < n) dinv[t] = rsqrtf(deg[t] + 1.0f);
}

// ---------------------------------------------------------------------------
// H = X @ W  via V_WMMA_F32_16X16X4_F32.
// Block = 128 threads = 4 waves. Each block computes a 16x64 tile of H:
// wave w owns output col-tile [16w, 16w+16). K=64 -> 16 chained WMMAs.
// A-tile (16x64, 4KB) and W (64x64, 16KB) staged in LDS.
// ---------------------------------------------------------------------------
__global__ void gemm_wmma_f32(const float* __restrict__ X, const float* __restrict__ W,
                              float* __restrict__ H) {
    __shared__ float As[16 * 64];
    __shared__ float Ws[64 * 64];

    const int row0 = blockIdx.x * 16;
    const int tid  = threadIdx.x;

    // cooperative loads (float4-coalesced)
    const float4* Xv = (const float4*)(X + (size_t)row0 * D);
    float4* Asv = (float4*)As;
    for (int i = tid; i < 16 * 64 / 4; i += 128) Asv[i] = Xv[i];
    const float4* Wv = (const float4*)W;
    float4* Wsv = (float4*)Ws;
    for (int i = tid; i < 64 * 64 / 4; i += 128) Wsv[i] = Wv[i];
    __syncthreads();

    const int wave = tid >> 5;       // 0..3 -> N col tile
    const int lane = tid & 31;
    const int half = lane >> 4;      // 0: K pair {0,1}; 1: K pair {2,3}
    const int l16  = lane & 15;
    const int n0   = wave * 16;

    v8f acc = {};
    #pragma unroll
    for (int kk = 0; kk < 16; ++kk) {
        const int kb = kk * 4 + half * 2;
        v2f a, b;
        // A 16x4 f32 layout: lane<16 -> row=l16, K=kb..kb+1 ; lane>=16 -> K shifted by 2
        a.x = As[l16 * 64 + kb + 0];
        a.y = As[l16 * 64 + kb + 1];
        // B 4x16 f32 (mirror of A): lane picks column n0+l16, VGPRs hold the K pair
        b.x = Ws[(kb + 0) * 64 + n0 + l16];
        b.y = Ws[(kb + 1) * 64 + n0 + l16];
        acc = __builtin_amdgcn_wmma_f32_16x16x4_f32(
            /*neg_a=*/false, a, /*neg_b=*/false, b,
            /*c_mod=*/(short)0, acc, /*reuse_a=*/false, /*reuse_b=*/false);
    }

    // C/D layout: VGPR r -> (M = r + 8*half, N = l16)
    float* Hrow = H + (size_t)row0 * D + n0;
    #pragma unroll
    for (int r = 0; r < 8; ++r)
        Hrow[(size_t)(r + 8 * half) * D + l16] = acc[r];
}

// ---------------------------------------------------------------------------
// edge scatter: agg[dst] += H[src] * dinv[src]*dinv[dst]
// 16 threads per edge, each handles 4 contiguous floats (float4 gather,
// 4 fp32 atomics). Destination tile is L2-resident (25.6MB << 192MB).
// ---------------------------------------------------------------------------
__global__ void scatter_kernel(const float* __restrict__ H,
                               const int* __restrict__ src, const int* __restrict__ dst,
                               const float* __restrict__ dinv,
                               float* __restrict__ agg, int n_edges) {
    int t = blockIdx.x * blockDim.x + threadIdx.x;
    int e = t >> 4;
    if (e >= n_edges) return;
    int part = (t & 15) * 4;
    int s = src[e], d = dst[e];
    float norm = dinv[s] * dinv[d];
    float4 hv = *(const float4*)(H + (size_t)s * D + part);
    float* out = agg + (size_t)d * D + part;
    atomicAdd(out + 0, hv.x * norm);
    atomicAdd(out + 1, hv.y * norm);
    atomicAdd(out + 2, hv.z * norm);
    atomicAdd(out + 3, hv.w * norm);
}

// ---------------------------------------------------------------------------
// epilogue: out = relu(agg + H * dinv^2 + b)   (dinv^2 == 1/deg self-loop)
// ---------------------------------------------------------------------------
__global__ void finalize_kernel(const float* __restrict__ agg, const float* __restrict__ H,
                                const float* __restrict__ dinv, const float* __restrict__ b,
                                float* __restrict__ out, int n_nodes) {
    int t = blockIdx.x * blockDim.x + threadIdx.x;
    if (t >= n_nodes * D) return;
    int i = t >> 6;
    int c = t & 63;
    float di = dinv[i];
    float v = agg[t] + H[t] * di * di + b[c];
    out[t] = fmaxf(v, 0.0f);
}

// ---------------------------------------------------------------------------
extern "C" void kernel_launch(void* const* d_in, const int* in_sizes, int n_in,
                              void* d_out, int out_size, void* d_ws, size_t ws_size,
                              hipStream_t stream) {
    const float* feature = (const float*)d_in[0];
    const int*   edges   = (const int*)d_in[1];
    const float* W1      = (const float*)d_in[2];
    const float* b1      = (const float*)d_in[3];
    const float* W2      = (const float*)d_in[4];
    const float* b2      = (const float*)d_in[5];
    float* out = (float*)d_out;

    const int n_nodes = in_sizes[0] / D;          // 100000
    const int n_edges = in_sizes[1] / 2;          // 1600000
    const int* src = edges;
    const int* dst = edges + n_edges;

    // workspace layout: [deg N][dinv N][h N*64][agg N*64]
    float* deg  = (float*)d_ws;
    float* dinv = deg  + n_nodes;
    float* h    = dinv + n_nodes;
    float* agg  = h    + (size_t)n_nodes * D;

    const int nfeat = n_nodes * D;
    dim3 b256(256);

    // ---- degrees (shared by both layers) ----
    zero_f32<<<(n_nodes + 255) / 256, b256, 0, stream>>>(deg, n_nodes);
    degree_kernel<<<(n_edges + 255) / 256, b256, 0, stream>>>(dst, deg, n_edges);
    dinv_kernel<<<(n_nodes + 255) / 256, b256, 0, stream>>>(deg, dinv, n_nodes);

    // ---- layer 1 ----
    zero_f32<<<(nfeat + 255) / 256, b256, 0, stream>>>(agg, nfeat);
    gemm_wmma_f32<<<n_nodes / 16, dim3(128), 0, stream>>>(feature, W1, h);
    scatter_kernel<<<(n_edges * 16 + 255) / 256, b256, 0, stream>>>(h, src, dst, dinv, agg, n_edges);
    // relu(agg + h/deg + b1) written in-place into agg -> becomes layer-2 input x1
    finalize_kernel<<<(nfeat + 255) / 256, b256, 0, stream>>>(agg, h, dinv, b1, agg, n_nodes);

    // ---- layer 2 ----
    gemm_wmma_f32<<<n_nodes / 16, dim3(128), 0, stream>>>(agg, W2, h);
    zero_f32<<<(nfeat + 255) / 256, b256, 0, stream>>>(agg, nfeat);
    scatter_kernel<<<(n_edges * 16 + 255) / 256, b256, 0, stream>>>(h, src, dst, dinv, agg, n_edges);
    finalize_kernel<<<(nfeat + 255) / 256, b256, 0, stream>>>(agg, h, dinv, b2, out, n_nodes);
}